// det_nlx_64287070487121
// MI455X (gfx1250) — compile-verified
//
#include <hip/hip_runtime.h>

#define NGI   13
#define NLOC  10
#define BATCH 262144

typedef __attribute__((ext_vector_type(2))) float v2f;
typedef __attribute__((ext_vector_type(8))) float v8f;

__global__ __launch_bounds__(256) void det_nlx_wmma_kernel(
    const float* __restrict__ x_loc,   // (B, 2, NLOC)
    const float* __restrict__ weight,  // (NGI)
    const float* __restrict__ w11,     // (NGI, NLOC)
    const float* __restrict__ w12,
    const float* __restrict__ w21,
    const float* __restrict__ w22,
    const float* __restrict__ nlx,     // (2, NLOC, NGI)
    float* __restrict__ nx,            // (B, 2, NGI, NLOC)
    float* __restrict__ detwei)        // (B, NGI)
{
    const int lane = threadIdx.x & 31;
    const int wave = threadIdx.x >> 5;
    const int tileBase = (blockIdx.x * 8 + wave) * 16;   // 16 batch rows per wave
    const int row = lane & 15;                           // A-matrix M index
    const int hi  = lane >> 4;                           // lane-half selects K pair
    const int g   = row;                                 // C/D N index owned by lane

    // ---------------- A fragments: x and y rows, K = 10 padded to 12 ----------------
    // A 16x4 f32 layout: lane(M=lane%16): VGPR0 = K=2*hi, VGPR1 = K=2*hi+1 (+4 per step)
    // Zero-padding via mask-multiply on clamped in-bounds loads (no divergent branches).
    const float* xrow = x_loc + (size_t)(tileBase + row) * (2 * NLOC);
    v2f ax[3], ay[3];
#pragma unroll
    for (int s = 0; s < 3; ++s) {
        const int k0 = 4 * s + 2 * hi;
        const int k1 = k0 + 1;
        const float m0 = (k0 < NLOC) ? 1.0f : 0.0f;
        const float m1 = (k1 < NLOC) ? 1.0f : 0.0f;
        const int c0 = (k0 < NLOC) ? k0 : (NLOC - 1);   // clamped, always in-bounds
        const int c1 = (k1 < NLOC) ? k1 : (NLOC - 1);
        ax[s][0] = xrow[c0] * m0;        ax[s][1] = xrow[c1] * m1;
        ay[s][0] = xrow[NLOC + c0] * m0; ay[s][1] = xrow[NLOC + c1] * m1;
    }

    // ---------------- B fragments: B[k][g] = w[g*NLOC + k], zero padded ----------------
    // B 4x16 layout mirrors A: lane(N=lane%16): VGPR0 = K=2*hi, VGPR1 = K=2*hi+1
    v2f b11[3], b12[3], b21[3], b22[3];
    {
        const float gm = (g < NGI) ? 1.0f : 0.0f;
        const int gc = (g < NGI) ? g : (NGI - 1);
#pragma unroll
        for (int s = 0; s < 3; ++s) {
            const int k0 = 4 * s + 2 * hi;
            const int k1 = k0 + 1;
            const float m0 = gm * ((k0 < NLOC) ? 1.0f : 0.0f);
            const float m1 = gm * ((k1 < NLOC) ? 1.0f : 0.0f);
            const int i0 = gc * NLOC + ((k0 < NLOC) ? k0 : (NLOC - 1));
            const int i1 = gc * NLOC + ((k1 < NLOC) ? k1 : (NLOC - 1));
            b11[s][0] = w11[i0] * m0;  b11[s][1] = w11[i1] * m1;
            b12[s][0] = w12[i0] * m0;  b12[s][1] = w12[i1] * m1;
            b21[s][0] = w21[i0] * m0;  b21[s][1] = w21[i1] * m1;
            b22[s][0] = w22[i0] * m0;  b22[s][1] = w22[i1] * m1;
        }
    }

    // ---------------- j = A x B via V_WMMA_F32_16X16X4_F32, 3 K-steps ----------------
    v8f j11 = {}, j12 = {}, j21 = {}, j22 = {};
#pragma unroll
    for (int s = 0; s < 3; ++s) {
        j11 = __builtin_amdgcn_wmma_f32_16x16x4_f32(false, ax[s], false, b11[s],
                                                    (short)0, j11, false, false);
        j12 = __builtin_amdgcn_wmma_f32_16x16x4_f32(false, ay[s], false, b12[s],
                                                    (short)0, j12, false, false);
        j21 = __builtin_amdgcn_wmma_f32_16x16x4_f32(false, ax[s], false, b21[s],
                                                    (short)0, j21, false, false);
        j22 = __builtin_amdgcn_wmma_f32_16x16x4_f32(false, ay[s], false, b22[s],
                                                    (short)0, j22, false, false);
    }

    // ---------------- epilogue: this lane owns g = lane%16, rows v + 8*hi ----------------
    if (g < NGI) {
        const float wg = weight[g];
        float a1[NLOC], a2[NLOC];
#pragma unroll
        for (int n = 0; n < NLOC; ++n) {
            a1[n] = nlx[n * NGI + g];               // nlx[0].T[g][n]
            a2[n] = nlx[NLOC * NGI + n * NGI + g];  // nlx[1].T[g][n]
        }
#pragma unroll
        for (int v = 0; v < 8; ++v) {
            const int b = tileBase + v + 8 * hi;    // C/D layout: M = v + 8*(lane/16)
            const float J11 = j11[v], J12 = j12[v], J21 = j21[v], J22 = j22[v];
            const float det = J11 * J22 - J21 * J12;
            const float invdet = 1.0f / det;
            detwei[(size_t)b * NGI + g] = fabsf(det) * wg;
            const float i11 =  J22 * invdet, i12 = -J12 * invdet;
            const float i21 = -J21 * invdet, i22 =  J11 * invdet;
            float* __restrict__ out1 = nx + (((size_t)b * 2 + 0) * NGI + g) * NLOC;
            float* __restrict__ out2 = nx + (((size_t)b * 2 + 1) * NGI + g) * NLOC;
#pragma unroll
            for (int n = 0; n < NLOC; n += 2) {     // rows are 40B-aligned -> b64 stores
                float2 o1, o2;
                o1.x = i11 * a1[n]     + i12 * a2[n];
                o1.y = i11 * a1[n + 1] + i12 * a2[n + 1];
                o2.x = i21 * a1[n]     + i22 * a2[n];
                o2.y = i21 * a1[n + 1] + i22 * a2[n + 1];
                *(float2*)(out1 + n) = o1;
                *(float2*)(out2 + n) = o2;
            }
        }
    }
}

extern "C" void kernel_launch(void* const* d_in, const int* in_sizes, int n_in,
                              void* d_out, int out_size, void* d_ws, size_t ws_size,
                              hipStream_t stream) {
    (void)in_sizes; (void)n_in; (void)out_size; (void)d_ws; (void)ws_size;
    const float* x_loc  = (const float*)d_in[0];
    const float* weight = (const float*)d_in[1];
    const float* w11    = (const float*)d_in[2];
    const float* w12    = (const float*)d_in[3];
    const float* w21    = (const float*)d_in[4];
    const float* w22    = (const float*)d_in[5];
    const float* nlx    = (const float*)d_in[6];

    float* nx     = (float*)d_out;                              // (B,2,NGI,NLOC) first
    float* detwei = nx + (size_t)BATCH * 2 * NGI * NLOC;        // then (B,NGI)

    const dim3 grid(BATCH / 128);   // 8 waves/block * 16 batch rows/wave
    const dim3 block(256);
    hipLaunchKernelGGL(det_nlx_wmma_kernel, grid, block, 0, stream,
                       x_loc, weight, w11, w12, w21, w22, nlx, nx, detwei);
}